// MultiheadAttention_3667902070967
// MI455X (gfx1250) — compile-verified
//
#include <hip/hip_runtime.h>
#include <math.h>

typedef float v2f __attribute__((ext_vector_type(2)));
typedef float v8f __attribute__((ext_vector_type(8)));

#define WMMA_F32(a, b, c) \
  __builtin_amdgcn_wmma_f32_16x16x4_f32(false, (a), false, (b), (short)0, (c), false, false)

#define SB 4
#define SS 512
#define SE 1024
#define SH 16
#define SDH 64
#define SE3 3072
#define SCALE 0.125f

__device__ __forceinline__ v2f vmax0(v2f a) {
  v2f r; r.x = fmaxf(a.x, 0.0f); r.y = fmaxf(a.y, 0.0f); return r;
}
__device__ __forceinline__ v2f vmin0(v2f a) {
  v2f r; r.x = fminf(a.x, 0.0f); r.y = fminf(a.y, 0.0f); return r;
}

// A fragment (16x4, row-major source, contiguous K pair): tp = &A[m0][k0]
__device__ __forceinline__ v2f load_a_frag(const float* __restrict__ tp, int ld, int lane) {
  int m = lane & 15, kh = lane >> 4;
  const float* p = tp + (long)m * ld + 2 * kh;
  v2f r; r.x = p[0]; r.y = p[1]; return r;
}

// B fragment (4x16, row-major source B[k][n]): tp = &B[k0][n0]
__device__ __forceinline__ v2f load_b_frag(const float* __restrict__ tp, int ld, int lane) {
  int n = lane & 15, kh = lane >> 4;
  const float* p = tp + (long)(2 * kh) * ld + n;
  v2f r; r.x = p[0]; r.y = p[ld]; return r;
}

// ---------------------------------------------------------------------------
// IBP linear:  Yv = Av@W + b ; Yl = Al@W+ + Au@W- + b ; Yu = Au@W+ + Al@W- + b
// One wave computes a 16x64 tile (4 n-subtiles). 3 accumulators per subtile.
// ---------------------------------------------------------------------------
__global__ __launch_bounds__(256) void ibp_linear_kernel(
    const float* __restrict__ Av, const float* __restrict__ Al, const float* __restrict__ Au,
    const float* __restrict__ W, const float* __restrict__ bias,
    float* __restrict__ Yv, float* __restrict__ Yl, float* __restrict__ Yu,
    int M, int N, int K) {
  const int lane = threadIdx.x & 31;
  const int wid = (blockIdx.x * blockDim.x + threadIdx.x) >> 5;
  const int ntiles = N >> 6;
  const int mt = wid / ntiles;
  const int nt = wid % ntiles;
  const int m0 = mt << 4;
  const int n0 = nt << 6;
  if (m0 >= M) return;

  v8f accV[4] = {}, accL[4] = {}, accU[4] = {};

  for (int k = 0; k < K; k += 4) {
    v2f aV = load_a_frag(Av + (long)m0 * K + k, K, lane);
    v2f aL = load_a_frag(Al + (long)m0 * K + k, K, lane);
    v2f aU = load_a_frag(Au + (long)m0 * K + k, K, lane);
    const float* wrow = W + (long)k * N + n0;
#pragma unroll
    for (int s = 0; s < 4; s++) {
      v2f w = load_b_frag(wrow + 16 * s, N, lane);
      v2f wP = vmax0(w), wN = vmin0(w);
      accV[s] = WMMA_F32(aV, w, accV[s]);
      accL[s] = WMMA_F32(aL, wP, accL[s]);
      accL[s] = WMMA_F32(aU, wN, accL[s]);
      accU[s] = WMMA_F32(aU, wP, accU[s]);
      accU[s] = WMMA_F32(aL, wN, accU[s]);
    }
  }

  const int ncol = lane & 15, kh = lane >> 4;
#pragma unroll
  for (int s = 0; s < 4; s++) {
#pragma unroll
    for (int r = 0; r < 8; r++) {
      long row = m0 + r + 8 * kh;
      int col = n0 + 16 * s + ncol;
      float bv = bias[col];
      Yv[row * N + col] = accV[s][r] + bv;
      Yl[row * N + col] = accL[s][r] + bv;
      Yu[row * N + col] = accU[s][r] + bv;
    }
  }
}

// ---------------------------------------------------------------------------
// Fused IBP attention for one (b, h, 16-row query tile).
// Block = 128 threads (4 waves). LDS holds 3 x [16 x 512] scores/probs.
// ---------------------------------------------------------------------------
__global__ __launch_bounds__(128) void ibp_attn_kernel(
    const float* __restrict__ QKVv, const float* __restrict__ QKVl, const float* __restrict__ QKVu,
    float* __restrict__ Ov, float* __restrict__ Ol, float* __restrict__ Ou) {
  extern __shared__ float smem[];
  float* sV = smem;                 // 16*512
  float* sL = sV + 16 * SS;
  float* sU = sL + 16 * SS;
  float* red = sU + 16 * SS;        // 3 * 128

  const int tid = threadIdx.x;
  const int lane = tid & 31;
  const int w = tid >> 5;

  const int bi = blockIdx.x;
  const int mt = bi & 31;           // S/16 tiles
  const int h = (bi >> 5) & 15;
  const int b = bi >> 9;
  const int m0 = mt << 4;

  const long base = (long)b * SS * SE3;
  const float* Qv = QKVv + base + h * SDH;
  const float* Ql = QKVl + base + h * SDH;
  const float* Qu = QKVu + base + h * SDH;
  const float* Kv = QKVv + base + SE + h * SDH;
  const float* Kl = QKVl + base + SE + h * SDH;
  const float* Ku = QKVu + base + SE + h * SDH;
  const float* Vv = QKVv + base + 2 * SE + h * SDH;
  const float* Vl = QKVl + base + 2 * SE + h * SDH;
  const float* Vu = QKVu + base + 2 * SE + h * SDH;

  // ---- Phase 1: scores s = (q*SCALE) @ k^T (interval x interval) ----------
  v2f qv[16], ql[16], qu[16];
#pragma unroll
  for (int kk = 0; kk < 16; kk++) {
    v2f a;
    a = load_a_frag(Qv + (long)m0 * SE3 + 4 * kk, SE3, lane); a.x *= SCALE; a.y *= SCALE; qv[kk] = a;
    a = load_a_frag(Ql + (long)m0 * SE3 + 4 * kk, SE3, lane); a.x *= SCALE; a.y *= SCALE; ql[kk] = a;
    a = load_a_frag(Qu + (long)m0 * SE3 + 4 * kk, SE3, lane); a.x *= SCALE; a.y *= SCALE; qu[kk] = a;
  }

  for (int i = 0; i < 8; i++) {
    const int nt = (w << 3) + i;
    const int n0 = nt << 4;
    v8f aV = {}, aL = {}, aU = {};
#pragma unroll 4
    for (int kk = 0; kk < 16; kk++) {
      const int k = 4 * kk;
      // K^T B-fragment == A-style contiguous load from row-major K
      v2f kv = load_a_frag(Kv + (long)n0 * SE3 + k, SE3, lane);
      v2f kl = load_a_frag(Kl + (long)n0 * SE3 + k, SE3, lane);
      v2f ku = load_a_frag(Ku + (long)n0 * SE3 + k, SE3, lane);
      v2f pql = vmax0(ql[kk]), nql = vmin0(ql[kk]);
      v2f pqu = vmax0(qu[kk]), nqu = vmin0(qu[kk]);
      v2f pkl = vmax0(kl), nkl = vmin0(kl);
      v2f pku = vmax0(ku), nku = vmin0(ku);
      aV = WMMA_F32(qv[kk], kv, aV);
      aL = WMMA_F32(pql, pkl, aL);
      aL = WMMA_F32(pqu, nkl, aL);
      aL = WMMA_F32(nql, pku, aL);
      aL = WMMA_F32(nqu, nku, aL);
      aU = WMMA_F32(pqu, pku, aU);
      aU = WMMA_F32(pql, nku, aU);
      aU = WMMA_F32(nqu, pkl, aU);
      aU = WMMA_F32(nql, nkl, aU);
    }
    const int ncol = lane & 15, kh = lane >> 4;
#pragma unroll
    for (int r = 0; r < 8; r++) {
      int row = r + 8 * kh;
      int col = n0 + ncol;
      sV[row * SS + col] = aV[r];
      sL[row * SS + col] = aL[r];
      sU[row * SS + col] = aU[r];
    }
  }
  __syncthreads();

  // ---- Phase 2: IBP softmax over each row of length 512 -------------------
  {
    const int r = tid >> 3;     // 0..15
    const int seg = tid & 7;    // 0..7 -> 64 cols each
    const int c0 = seg * 64;
    float mv = -INFINITY, mu = -INFINITY;
    for (int c = 0; c < 64; c++) {
      mv = fmaxf(mv, sV[r * SS + c0 + c]);
      mu = fmaxf(mu, sU[r * SS + c0 + c]);
    }
    red[r * 8 + seg] = mv;
    red[128 + r * 8 + seg] = mu;
    __syncthreads();
    mv = red[r * 8]; mu = red[128 + r * 8];
    for (int j = 1; j < 8; j++) {
      mv = fmaxf(mv, red[r * 8 + j]);
      mu = fmaxf(mu, red[128 + r * 8 + j]);
    }
    __syncthreads();
    float sv = 0.0f, sl = 0.0f, su = 0.0f;
    for (int c = 0; c < 64; c++) {
      sv += __expf(sV[r * SS + c0 + c] - mv);
      sl += __expf(sL[r * SS + c0 + c] - mu);
      su += __expf(sU[r * SS + c0 + c] - mu);
    }
    red[r * 8 + seg] = sv;
    red[128 + r * 8 + seg] = sl;
    red[256 + r * 8 + seg] = su;
    __syncthreads();
    float Svs = 0.0f, Sls = 0.0f, Sus = 0.0f;
    for (int j = 0; j < 8; j++) {
      Svs += red[r * 8 + j];
      Sls += red[128 + r * 8 + j];
      Sus += red[256 + r * 8 + j];
    }
    const float invSv = 1.0f / Svs;
    for (int c = 0; c < 64; c++) {
      const int idx = r * SS + c0 + c;
      float ev = __expf(sV[idx] - mv);
      float el = __expf(sL[idx] - mu);
      float eu = __expf(sU[idx] - mu);
      float pv = ev * invSv;
      float pl = el / (Sus - eu + el);
      float pu = eu / (Sls - el + eu);
      pl = fminf(fmaxf(pl, 0.0f), 1.0f);
      pu = fminf(fmaxf(pu, 0.0f), 1.0f);
      sV[idx] = pv;
      sL[idx] = pl;
      sU[idx] = pu;
    }
  }
  __syncthreads();

  // ---- Phase 3: o = p @ v. pl,pu >= 0 so only V needs clamping ------------
  {
    const int n0 = w * 16;  // 16 cols of DH per wave
    v8f oV = {}, oL = {}, oU = {};
    const int m = lane & 15, kh = lane >> 4;
    for (int k = 0; k < SS; k += 4) {
      const int off = m * SS + k + 2 * kh;
      v2f aPv, aPl, aPu;
      aPv.x = sV[off]; aPv.y = sV[off + 1];
      aPl.x = sL[off]; aPl.y = sL[off + 1];
      aPu.x = sU[off]; aPu.y = sU[off + 1];
      v2f bv = load_b_frag(Vv + (long)k * SE3 + n0, SE3, lane);
      v2f bl = load_b_frag(Vl + (long)k * SE3 + n0, SE3, lane);
      v2f bu = load_b_frag(Vu + (long)k * SE3 + n0, SE3, lane);
      v2f pbl = vmax0(bl), nbl = vmin0(bl);
      v2f pbu = vmax0(bu), nbu = vmin0(bu);
      oV = WMMA_F32(aPv, bv, oV);
      oL = WMMA_F32(aPl, pbl, oL);
      oL = WMMA_F32(aPu, nbl, oL);
      oU = WMMA_F32(aPu, pbu, oU);
      oU = WMMA_F32(aPl, nbu, oU);
    }
    const int ncol = lane & 15;
#pragma unroll
    for (int r = 0; r < 8; r++) {
      long row = (long)b * SS + m0 + r + 8 * kh;
      int col = h * SDH + n0 + ncol;
      Ov[row * SE + col] = oV[r];
      Ol[row * SE + col] = oL[r];
      Ou[row * SE + col] = oU[r];
    }
  }
}

extern "C" void kernel_launch(void* const* d_in, const int* in_sizes, int n_in,
                              void* d_out, int out_size, void* d_ws, size_t ws_size,
                              hipStream_t stream) {
  const float* x_val = (const float*)d_in[0];
  const float* x_lb = (const float*)d_in[1];
  const float* x_ub = (const float*)d_in[2];
  const float* Wi = (const float*)d_in[3];
  const float* bi = (const float*)d_in[4];
  const float* Wo = (const float*)d_in[5];
  const float* bo = (const float*)d_in[6];
  float* out = (float*)d_out;

  float* ws = (float*)d_ws;
  const long QKVN = (long)SB * SS * SE3;  // 6.29M floats each
  const long ON = (long)SB * SS * SE;     // 2.10M floats each
  float* qkv_v = ws;
  float* qkv_l = ws + QKVN;
  float* qkv_u = ws + 2 * QKVN;
  float* o_v = ws + 3 * QKVN;
  float* o_l = o_v + ON;
  float* o_u = o_l + ON;

  // 1) in_proj IBP linear: [2048,1024] x [1024,3072]
  {
    const int M = SB * SS, N = SE3, K = SE;
    const int waves = (M / 16) * (N / 64);
    ibp_linear_kernel<<<dim3(waves / 8), 256, 0, stream>>>(
        x_val, x_lb, x_ub, Wi, bi, qkv_v, qkv_l, qkv_u, M, N, K);
  }

  // 2) fused IBP attention: one block per (b, h, 16-row query tile)
  {
    const size_t sm = (size_t)(3 * 16 * SS + 3 * 128) * sizeof(float);  // ~97.5 KB
    ibp_attn_kernel<<<dim3(SB * SH * (SS / 16)), 128, sm, stream>>>(
        qkv_v, qkv_l, qkv_u, o_v, o_l, o_u);
  }

  // 3) out_proj IBP linear: [2048,1024] x [1024,1024] -> stacked [3,B,S,E]
  {
    const int M = SB * SS, N = SE, K = SE;
    const int waves = (M / 16) * (N / 64);
    ibp_linear_kernel<<<dim3(waves / 8), 256, 0, stream>>>(
        o_v, o_l, o_u, Wo, bo, out, out + (long)SB * SS * SE, out + 2L * SB * SS * SE, M, N, K);
  }
}